// LIFNode_25202868093202
// MI455X (gfx1250) — compile-verified
//
#include <hip/hip_runtime.h>
#include <stdint.h>

// ---------------------------------------------------------------------------
// MI455X (gfx1250) implementation.
//
// Workload is pure HBM streaming (~572 MB total traffic, ~25 us roofline at
// 23.3 TB/s); no matmul -> WMMA inapplicable. We use the CDNA5 async
// global->LDS path (GLOBAL_LOAD_ASYNC_TO_LDS_B128 + s_wait_asynccnt) to
// double-buffer the x_seq stream for the sequential-in-t LIF scan.
// ---------------------------------------------------------------------------

typedef int v4i __attribute__((ext_vector_type(4)));
typedef __attribute__((address_space(1))) v4i gas_v4i;  // global (AS1) int4
typedef __attribute__((address_space(3))) v4i las_v4i;  // LDS (AS3) int4

// Async copy of 16 bytes (per lane) from global to LDS. Tracked by ASYNCcnt.
__device__ __forceinline__ void async_copy_b128(const void* gsrc, void* lds_dst) {
#if defined(__has_builtin) && __has_builtin(__builtin_amdgcn_global_load_async_to_lds_b128)
  // Signature (from clang diagnostic): (int4 AS1* gsrc, int4 AS3* ldst, imm, imm cpol)
  __builtin_amdgcn_global_load_async_to_lds_b128(
      (gas_v4i*)(uintptr_t)gsrc,
      (las_v4i*)(uint32_t)(uintptr_t)lds_dst,  // low 32 bits of flat LDS addr == LDS offset
      0, 0);
#else
  // Fallback: raw CDNA5 encoding, SADDR = off (64-bit vaddr in VGPR pair),
  // VDST = LDS byte offset.
  uint32_t lds_off = (uint32_t)(uintptr_t)lds_dst;
  asm volatile("global_load_async_to_lds_b128 %0, %1, off"
               :
               : "v"(lds_off), "v"((uint64_t)(uintptr_t)gsrc)
               : "memory");
#endif
}

__device__ __forceinline__ void wait_async_le1() {
  asm volatile("s_wait_asynccnt 1" ::: "memory");
}
__device__ __forceinline__ void wait_async_le0() {
  asm volatile("s_wait_asynccnt 0" ::: "memory");
}

// ---------------------------------------------------------------------------
// Kernel A: multi-step LIF scan. TAU=2, V_TH=1, V_RESET=0, hard reset.
// One thread owns 4 contiguous d-elements across all T steps.
// Each lane async-stages its own 16B x-slice into its own LDS slot
// (double buffered), so no cross-lane sharing -> no barriers, only
// s_wait_asynccnt. Loads retire in order, so "<=1" frees the older buffer.
// ---------------------------------------------------------------------------
__global__ __launch_bounds__(256) void lif_scan_kernel(
    const float* __restrict__ x, float* __restrict__ out, long long nd4, int T) {
  __shared__ float4 tile[2][256];  // 8 KB of the 320 KB WGP LDS

  const int tid = threadIdx.x;
  const long long base = (long long)blockIdx.x * 256 + tid;  // float4 index
  if (base >= nd4) return;

  const float4* __restrict__ x4 = (const float4*)x;
  float4* __restrict__ o4 = (float4*)out;

  // Prefetch t = 0.
  async_copy_b128(&x4[base], &tile[0][tid]);

  float4 v;
  v.x = v.y = v.z = v.w = 0.0f;

  for (int t = 0; t < T; ++t) {
    if (t + 1 < T) {
      async_copy_b128(&x4[(long long)(t + 1) * nd4 + base], &tile[(t + 1) & 1][tid]);
      wait_async_le1();  // buffer for step t is complete
    } else {
      wait_async_le0();
    }

    const float4 xv = tile[t & 1][tid];  // ds_load_b128
    float4 s;
    {
      float h = v.x + (xv.x - v.x) * 0.5f;  // charge: v + (x - v)/tau
      s.x = (h >= 1.0f) ? 1.0f : 0.0f;      // fire
      v.x = (h >= 1.0f) ? 0.0f : h;         // hard reset (V_RESET = 0)
    }
    {
      float h = v.y + (xv.y - v.y) * 0.5f;
      s.y = (h >= 1.0f) ? 1.0f : 0.0f;
      v.y = (h >= 1.0f) ? 0.0f : h;
    }
    {
      float h = v.z + (xv.z - v.z) * 0.5f;
      s.z = (h >= 1.0f) ? 1.0f : 0.0f;
      v.z = (h >= 1.0f) ? 0.0f : h;
    }
    {
      float h = v.w + (xv.w - v.w) * 0.5f;
      s.w = (h >= 1.0f) ? 1.0f : 0.0f;
      v.w = (h >= 1.0f) ? 0.0f : h;
    }
    o4[(long long)t * nd4 + base] = s;  // global_store_b128
  }
}

// ---------------------------------------------------------------------------
// Kernel B: Lorentz exponential map, one wave32 per row of D=512.
// inner = -v0^2 + sum_{i>=1} vi^2 = (sum_i vi^2) - 2*v0^2
// out   = cosh(n)*z + (sinh(n)/n)*v,  n = sqrt(max(inner, 1e-6))
// ---------------------------------------------------------------------------
__global__ __launch_bounds__(256) void lorentz_expmap_kernel(
    const float* __restrict__ v, const float* __restrict__ z,
    float* __restrict__ out, long long nrows) {
  const int lane = threadIdx.x & 31;
  const long long row = (long long)blockIdx.x * 8 + (threadIdx.x >> 5);
  if (row >= nrows) return;

  const float4* __restrict__ v4 = (const float4*)(v + row * 512);
  const float4* __restrict__ z4 = (const float4*)(z + row * 512);
  float4* __restrict__ o4 = (float4*)(out + row * 512);

  // 512 floats / 32 lanes = 4 float4 per lane, coalesced b128 loads.
  float4 a0 = v4[lane];
  float4 a1 = v4[lane + 32];
  float4 a2 = v4[lane + 64];
  float4 a3 = v4[lane + 96];

  float ss = a0.x * a0.x + a0.y * a0.y + a0.z * a0.z + a0.w * a0.w;
  ss += a1.x * a1.x + a1.y * a1.y + a1.z * a1.z + a1.w * a1.w;
  ss += a2.x * a2.x + a2.y * a2.y + a2.z * a2.z + a2.w * a2.w;
  ss += a3.x * a3.x + a3.y * a3.y + a3.z * a3.z + a3.w * a3.w;

  // Full wave32 butterfly reduction.
#pragma unroll
  for (int off = 16; off > 0; off >>= 1) ss += __shfl_xor(ss, off, 32);

  // v0 lives in lane 0's a0.x (d == 0); broadcast it.
  const float v0 = __shfl(a0.x, 0, 32);
  const float inner = ss - 2.0f * v0 * v0;
  const float nrm = sqrtf(fmaxf(inner, 1e-6f));

  const float e = expf(nrm);
  const float ei = 1.0f / e;
  const float ch = 0.5f * (e + ei);           // cosh
  const float shn = (0.5f * (e - ei)) / nrm;  // sinh / |v|

  float4 r;
  float4 b = z4[lane];
  r.x = ch * b.x + shn * a0.x; r.y = ch * b.y + shn * a0.y;
  r.z = ch * b.z + shn * a0.z; r.w = ch * b.w + shn * a0.w;
  o4[lane] = r;
  b = z4[lane + 32];
  r.x = ch * b.x + shn * a1.x; r.y = ch * b.y + shn * a1.y;
  r.z = ch * b.z + shn * a1.z; r.w = ch * b.w + shn * a1.w;
  o4[lane + 32] = r;
  b = z4[lane + 64];
  r.x = ch * b.x + shn * a2.x; r.y = ch * b.y + shn * a2.y;
  r.z = ch * b.z + shn * a2.z; r.w = ch * b.w + shn * a2.w;
  o4[lane + 64] = r;
  b = z4[lane + 96];
  r.x = ch * b.x + shn * a3.x; r.y = ch * b.y + shn * a3.y;
  r.z = ch * b.z + shn * a3.z; r.w = ch * b.w + shn * a3.w;
  o4[lane + 96] = r;
}

// ---------------------------------------------------------------------------
// Launch. Inputs: x_seq [T,N,D] f32, v_seq [N,D] f32, z_seq [N,D] f32.
// Output: spikes [T,N,D] f32 followed by z_out [N,D] f32 (tuple flattened).
// ---------------------------------------------------------------------------
extern "C" void kernel_launch(void* const* d_in, const int* in_sizes, int n_in,
                              void* d_out, int out_size, void* d_ws, size_t ws_size,
                              hipStream_t stream) {
  const float* x = (const float*)d_in[0];
  const float* v = (const float*)d_in[1];
  const float* z = (const float*)d_in[2];
  float* out = (float*)d_out;

  const long long ND = (long long)in_sizes[1];          // N*D = 4,194,304
  const int T = (int)((long long)in_sizes[0] / ND);     // 16
  const long long nd4 = ND / 4;                         // float4 elements per step
  const long long nrows = ND / 512;                     // N (D = 512 per reference)

  const unsigned lif_blocks = (unsigned)((nd4 + 255) / 256);  // 4096
  lif_scan_kernel<<<lif_blocks, 256, 0, stream>>>(x, out, nd4, T);

  float* zout = out + (long long)T * ND;
  const unsigned lz_blocks = (unsigned)((nrows + 7) / 8);  // 1024
  lorentz_expmap_kernel<<<lz_blocks, 256, 0, stream>>>(v, z, zout, nrows);
}